// BlockWiseEmbedding_72335839199518
// MI455X (gfx1250) — compile-verified
//
#include <hip/hip_runtime.h>
#include <stdint.h>

// Blockwise embedding gather for MI455X (gfx1250).
// HBM-bound copy (~268 MB, 0 FLOPs; ~11.5us floor @ 23.3 TB/s). Per 2 KB row:
//   4x global_load_async_to_lds_b128   (HBM -> LDS, ASYNCcnt)
//   s_wait_asynccnt 0
//   4x global_store_async_from_lds_b128 th:TH_STORE_NT (LDS -> HBM)
// Stores are non-temporal: output is write-once, so keep the 192 MB L2 full
// of table rows (which get re-referenced) instead of dead output lines.
// GVS addressing; INST_OFFSET adds to BOTH global and LDS addresses, so all
// per-token/per-chunk placement lives in instruction immediates.

#define DIM 512
#define ROW_BYTES (DIM * 4)          // 2048 bytes per embedding row
#define WAVES_PER_BLOCK 4
#define TOK_PER_WAVE 4
#define THREADS (WAVES_PER_BLOCK * 32)
#define TOK_PER_BLOCK (WAVES_PER_BLOCK * TOK_PER_WAVE)

#define ASYNC_LD4(LDSV, VOFF, SROW, O0, O1, O2, O3)                            \
    do {                                                                       \
        asm volatile("global_load_async_to_lds_b128 %0, %1, %2 offset:" #O0    \
                     :: "v"(LDSV), "v"(VOFF), "s"(SROW) : "memory");           \
        asm volatile("global_load_async_to_lds_b128 %0, %1, %2 offset:" #O1    \
                     :: "v"(LDSV), "v"(VOFF), "s"(SROW) : "memory");           \
        asm volatile("global_load_async_to_lds_b128 %0, %1, %2 offset:" #O2    \
                     :: "v"(LDSV), "v"(VOFF), "s"(SROW) : "memory");           \
        asm volatile("global_load_async_to_lds_b128 %0, %1, %2 offset:" #O3    \
                     :: "v"(LDSV), "v"(VOFF), "s"(SROW) : "memory");           \
    } while (0)

#define ASYNC_ST4(VOFF, LDSV, DBASE, O0, O1, O2, O3)                           \
    do {                                                                       \
        asm volatile("global_store_async_from_lds_b128 %0, %1, %2 offset:" #O0 \
                     " th:TH_STORE_NT"                                         \
                     :: "v"(VOFF), "v"(LDSV), "s"(DBASE) : "memory");          \
        asm volatile("global_store_async_from_lds_b128 %0, %1, %2 offset:" #O1 \
                     " th:TH_STORE_NT"                                         \
                     :: "v"(VOFF), "v"(LDSV), "s"(DBASE) : "memory");          \
        asm volatile("global_store_async_from_lds_b128 %0, %1, %2 offset:" #O2 \
                     " th:TH_STORE_NT"                                         \
                     :: "v"(VOFF), "v"(LDSV), "s"(DBASE) : "memory");          \
        asm volatile("global_store_async_from_lds_b128 %0, %1, %2 offset:" #O3 \
                     " th:TH_STORE_NT"                                         \
                     :: "v"(VOFF), "v"(LDSV), "s"(DBASE) : "memory");          \
    } while (0)

__global__ __launch_bounds__(THREADS)
void BlockWiseEmbedding_gather_async(const int* __restrict__ src,
                                     const int* __restrict__ blk,
                                     const int* __restrict__ loc,
                                     const float* __restrict__ t0,
                                     const float* __restrict__ t1,
                                     const float* __restrict__ t2,
                                     const float* __restrict__ t3,
                                     float* __restrict__ out,
                                     int n_tokens)
{
    __shared__ float lds[WAVES_PER_BLOCK * TOK_PER_WAVE * DIM];   // 32 KB

    const int lane = threadIdx.x & 31;
    const int wave = __builtin_amdgcn_readfirstlane((int)(threadIdx.x >> 5));
    const int base_tok = (int)blockIdx.x * TOK_PER_BLOCK + wave * TOK_PER_WAVE;

    // One LDS-address VGPR serves all 32 async ops (placement is carried by
    // the instruction immediates, which offset LDS and global identically).
    const uint32_t ldsv =
        (uint32_t)(uintptr_t)(&lds[wave * TOK_PER_WAVE * DIM])
        + (uint32_t)lane * 16u;
    const uint32_t voff = (uint32_t)lane * 16u;   // global vaddr: lane slice

    // ---- Index chase, level by level (independent SMEM loads clause up,
    // one s_wait_kmcnt per level). Unsigned math avoids sign-extends. ----
    uint32_t stok[TOK_PER_WAVE], ids[TOK_PER_WAVE], bs[TOK_PER_WAVE], ls[TOK_PER_WAVE];
#pragma unroll
    for (int t = 0; t < TOK_PER_WAVE; ++t) {
        const int tok = base_tok + t;
        stok[t] = (tok < n_tokens) ? (uint32_t)tok : 0u;  // clamp: loads safe
    }
#pragma unroll
    for (int t = 0; t < TOK_PER_WAVE; ++t) ids[t] = (uint32_t)src[stok[t]];
#pragma unroll
    for (int t = 0; t < TOK_PER_WAVE; ++t) bs[t] = (uint32_t)blk[ids[t]];
#pragma unroll
    for (int t = 0; t < TOK_PER_WAVE; ++t) ls[t] = (uint32_t)loc[ids[t]];

    // ---- Branchless table mux (s_cselect_b64 x3) and row pointer biased by
    // -t*ROW_BYTES so the shared immediates line up for LDS and global. ----
    const char* srow[TOK_PER_WAVE];
#pragma unroll
    for (int t = 0; t < TOK_PER_WAVE; ++t) {
        const uint32_t b = bs[t];
        const uint64_t pa = (b & 1u) ? (uint64_t)(uintptr_t)t1
                                     : (uint64_t)(uintptr_t)t0;
        const uint64_t pb = (b & 1u) ? (uint64_t)(uintptr_t)t3
                                     : (uint64_t)(uintptr_t)t2;
        const uint64_t tp = (b & 2u) ? pb : pa;
        srow[t] = (const char*)(uintptr_t)(tp + (uint64_t)ls[t] * ROW_BYTES
                                              - (uint64_t)(t * ROW_BYTES));
    }

    // ---- Phase 1: 16 outstanding HBM->LDS async b128 loads. ----
    ASYNC_LD4(ldsv, voff, srow[0],    0,  512, 1024, 1536);
    ASYNC_LD4(ldsv, voff, srow[1], 2048, 2560, 3072, 3584);
    ASYNC_LD4(ldsv, voff, srow[2], 4096, 4608, 5120, 5632);
    ASYNC_LD4(ldsv, voff, srow[3], 6144, 6656, 7168, 7680);

    asm volatile("s_wait_asynccnt 0" ::: "memory");

    // ---- Phase 2: non-temporal async LDS->HBM stores. Destination rows are
    // contiguous: one scalar base, immediates carry the rest. S_ENDPGM's
    // implicit wait-idle covers completion. ----
    const float* dbase = out + (size_t)base_tok * DIM;
    if (base_tok + 0 < n_tokens) ASYNC_ST4(voff, ldsv, dbase,    0,  512, 1024, 1536);
    if (base_tok + 1 < n_tokens) ASYNC_ST4(voff, ldsv, dbase, 2048, 2560, 3072, 3584);
    if (base_tok + 2 < n_tokens) ASYNC_ST4(voff, ldsv, dbase, 4096, 4608, 5120, 5632);
    if (base_tok + 3 < n_tokens) ASYNC_ST4(voff, ldsv, dbase, 6144, 6656, 7168, 7680);
}

extern "C" void kernel_launch(void* const* d_in, const int* in_sizes, int n_in,
                              void* d_out, int out_size, void* d_ws, size_t ws_size,
                              hipStream_t stream) {
    // setup_inputs() order: src, block_assign, local_assign, table0..table3
    const int*   src = (const int*)d_in[0];
    const int*   blk = (const int*)d_in[1];
    const int*   loc = (const int*)d_in[2];
    const float* t0  = (const float*)d_in[3];
    const float* t1  = (const float*)d_in[4];
    const float* t2  = (const float*)d_in[5];
    const float* t3  = (const float*)d_in[6];
    float*       out = (float*)d_out;

    const int n_tokens = in_sizes[0];                 // B*S = 65536
    const int blocks = (n_tokens + TOK_PER_BLOCK - 1) / TOK_PER_BLOCK;

    BlockWiseEmbedding_gather_async<<<blocks, THREADS, 0, stream>>>(
        src, blk, loc, t0, t1, t2, t3, out, n_tokens);
}